// Top1Router_38508676776575
// MI455X (gfx1250) — compile-verified
//
#include <hip/hip_runtime.h>

// Problem constants (fixed by the reference harness)
#define S_TOKENS 8192
#define N_EXPERTS 64
#define CAP 160          // ceil(1.25 * 8192 / 64)
#define EK_THREADS 256   // 8 waves in the per-expert kernel
#define EK_WAVES 8
#define SEG (S_TOKENS / EK_WAVES)  // 1024 tokens per wave segment

typedef float v4f __attribute__((ext_vector_type(4)));
typedef int v4i __attribute__((ext_vector_type(4)));

#if __has_builtin(__builtin_amdgcn_global_load_async_to_lds_b128) && \
    __has_builtin(__builtin_amdgcn_s_wait_asynccnt)
#define HAVE_ASYNC_LDS 1
// Builtin signature (from hipcc diagnostic): first arg is v4i* in the global
// (__device__/AS1) address space; LDS arg is v4i* in shared (AS3).
typedef __attribute__((address_space(1))) v4i* as1_v4i;
typedef __attribute__((address_space(3))) v4i* as3_v4i;
#else
#define HAVE_ASYNC_LDS 0
#endif

__device__ __forceinline__ unsigned ballot32(bool p) {
#if __has_builtin(__builtin_amdgcn_ballot_w32)
  return __builtin_amdgcn_ballot_w32(p);
#else
  return (unsigned)__ballot(p ? 1 : 0);
#endif
}

// ---------------------------------------------------------------------------
// Kernel 1: zero-fill d_out with wide non-temporal stores (bandwidth floor).
// ---------------------------------------------------------------------------
__global__ __launch_bounds__(256) void zero_out_kernel(float* __restrict__ out,
                                                       long long total) {
  long long nvec = total >> 2;
  long long i = (long long)blockIdx.x * blockDim.x + threadIdx.x;
  long long stride = (long long)gridDim.x * blockDim.x;
  v4f z = (v4f)0.0f;
  v4f* o = (v4f*)out;
  for (; i < nvec; i += stride) {
    __builtin_nontemporal_store(z, o + i);
  }
  if (blockIdx.x == 0 && threadIdx.x == 0) {
    for (long long t = (nvec << 2); t < total; ++t) out[t] = 0.0f;
  }
}

// ---------------------------------------------------------------------------
// Kernel 2: per-token router. One wave32 per token, 2 experts per lane.
// Computes argmax (tie -> lowest index, matching jnp.argmax), the softmax
// probability AT the argmax (= 1/sum(exp(x-max))), and rand[s, argmax].
// ---------------------------------------------------------------------------
__global__ __launch_bounds__(256) void router_kernel(
    const float* __restrict__ x, const float* __restrict__ rnd,
    int* __restrict__ eidx, float* __restrict__ gate,
    float* __restrict__ randv) {
  const int tok = (blockIdx.x * blockDim.x + threadIdx.x) >> 5;
  const int lane = threadIdx.x & 31;
  if (tok >= S_TOKENS) return;

  const float2 v = *(const float2*)(x + (size_t)tok * N_EXPERTS + lane * 2);

  // local (value, index) argmax; tie prefers lower index
  float m;
  int mi;
  if (v.x >= v.y) { m = v.x; mi = lane * 2; }
  else            { m = v.y; mi = lane * 2 + 1; }

  // wave32 butterfly reduce: max value, tie -> min index
  for (int off = 16; off > 0; off >>= 1) {
    float om = __shfl_xor(m, off, 32);
    int omi = __shfl_xor(mi, off, 32);
    if (om > m || (om == m && omi < mi)) { m = om; mi = omi; }
  }

  float ssum = __expf(v.x - m) + __expf(v.y - m);
  for (int off = 16; off > 0; off >>= 1) ssum += __shfl_xor(ssum, off, 32);

  if (lane == 0) {
    eidx[tok] = mi;
    gate[tok] = 1.0f / ssum;  // probs[tok, argmax]
    randv[tok] = rnd[(size_t)tok * N_EXPERTS + mi];
  }
}

// ---------------------------------------------------------------------------
// Kernel 3: one workgroup (8 waves) per expert.
//   Phase A: stage expert_idx[8192] into LDS (async-to-LDS when available).
//   Phase B: per-wave counts, then ordered compaction (token order preserved
//            via ballot prefix sums -> cumsum positions come for free).
//   Phase C (wave 0): rare N>CAP case -> bitwise binary search for the
//            160th-largest rand among assigned tokens; stable tie-break
//            (lowest token index first) matches jax.lax.top_k.
//   Phase D (wave 0): scatter combine weight + mask bit; write exp_counts.
// ---------------------------------------------------------------------------
__global__ __launch_bounds__(EK_THREADS) void expert_kernel(
    const int* __restrict__ eidx, const float* __restrict__ gate,
    const float* __restrict__ randv, float* __restrict__ combine,
    float* __restrict__ maskout, float* __restrict__ counts) {
  const int e = blockIdx.x;
  const int tid = threadIdx.x;
  const int wave = tid >> 5;
  const int lane = tid & 31;
  const unsigned lt = (1u << lane) - 1u;

  __shared__ int s_eidx[S_TOKENS];           // 32 KB
  __shared__ float s_rv[S_TOKENS];           // 32 KB
  __shared__ unsigned short s_tid[S_TOKENS]; // 16 KB
  __shared__ int s_wcnt[EK_WAVES];

  // ---- Phase A: stage expert indices into LDS ----
#if HAVE_ASYNC_LDS
  {
    char* g = (char*)(const void*)eidx;
    char* l = (char*)s_eidx;
    for (int byte = tid * 16; byte < S_TOKENS * 4; byte += EK_THREADS * 16) {
      __builtin_amdgcn_global_load_async_to_lds_b128(
          (as1_v4i)(g + byte), (as3_v4i)(l + byte), 0, 0);
    }
    __builtin_amdgcn_s_wait_asynccnt(0);
  }
#else
  for (int i = tid; i < S_TOKENS; i += EK_THREADS) s_eidx[i] = eidx[i];
#endif
  __syncthreads();

  // ---- Phase B1: per-wave assigned counts over its 1024-token segment ----
  const int seg0 = wave * SEG;
  int cnt = 0;
  for (int b = 0; b < SEG; b += 32) {
    bool a = (s_eidx[seg0 + b + lane] == e);
    cnt += __popc(ballot32(a));
  }
  if (lane == 0) s_wcnt[wave] = cnt;
  __syncthreads();

  int base = 0;
  for (int w = 0; w < wave; ++w) base += s_wcnt[w];
  int N = 0;
  for (int w = 0; w < EK_WAVES; ++w) N += s_wcnt[w];

  // ---- Phase B2: ordered compaction (token-index order preserved) ----
  int off = base;
  for (int b = 0; b < SEG; b += 32) {
    const int tok = seg0 + b + lane;
    bool a = (s_eidx[tok] == e);
    unsigned m = ballot32(a);
    int pre = __popc(m & lt);
    if (a) {
      s_tid[off + pre] = (unsigned short)tok;
      s_rv[off + pre] = randv[tok];
    }
    off += __popc(m);
  }
  __syncthreads();

  // ---- Phases C & D on wave 0 only (no barriers below) ----
  if (wave != 0) return;

  bool need_thresh = (N > CAP);
  float tau = -1.0f;
  int ties_needed = 0;
  if (need_thresh) {
    // rand in [0,1): positive f32 bits compare like floats -> bit binary
    // search for smallest t with count(rv > t) < CAP, i.e. the CAP-th largest.
    unsigned lo = 0u, hi = 0x3F800000u;  // (0.0f, 1.0f]
    while (hi - lo > 1u) {
      unsigned mid = (lo + hi) >> 1;
      float t = __uint_as_float(mid);
      int c = 0;
      for (int i = lane; i < N; i += 32) c += (s_rv[i] > t) ? 1 : 0;
      for (int o = 16; o > 0; o >>= 1) c += __shfl_xor(c, o, 32);
      if (c >= CAP) lo = mid; else hi = mid;
    }
    tau = __uint_as_float(hi);
    int c = 0;
    for (int i = lane; i < N; i += 32) c += (s_rv[i] > tau) ? 1 : 0;
    for (int o = 16; o > 0; o >>= 1) c += __shfl_xor(c, o, 32);
    ties_needed = CAP - c;  // earliest-token ties at tau fill the remainder
  }

  // ---- Phase D: keep flags, loc = running kept count, scatter writes ----
  int kept = 0, ties = 0;
  for (int bse = 0; bse < N; bse += 32) {
    const int i = bse + lane;
    const bool valid = (i < N);
    const float rv = valid ? s_rv[i] : -2.0f;
    bool keep;
    if (!need_thresh) {
      keep = valid;
    } else {
      bool gt = valid && (rv > tau);
      bool tie = valid && (rv == tau);
      unsigned tmask = ballot32(tie);
      int tpre = __popc(tmask & lt);
      keep = gt || (tie && (ties + tpre) < ties_needed);
      ties += __popc(tmask);
    }
    unsigned kmask = ballot32(keep);
    int kpre = __popc(kmask & lt);
    if (keep) {
      const int loc = kept + kpre;  // == cumsum(dispatch)-1 at this token
      const int tok = (int)s_tid[i];
      const size_t o = ((size_t)tok * N_EXPERTS + e) * CAP + (size_t)loc;
      combine[o] = gate[tok];
      maskout[o] = 1.0f;
    }
    kept += __popc(kmask);
  }

  if (lane == 0) counts[e] = (float)N;  // exp_counts (ALL assigned, pre-drop)
}

// ---------------------------------------------------------------------------
extern "C" void kernel_launch(void* const* d_in, const int* in_sizes, int n_in,
                              void* d_out, int out_size, void* d_ws,
                              size_t ws_size, hipStream_t stream) {
  const float* x = (const float*)d_in[0];    // inputs [8192, 64] f32
  const float* rnd = (const float*)d_in[1];  // rand   [8192, 64] f32

  const size_t per_tensor = (size_t)S_TOKENS * N_EXPERTS * CAP;  // 83,886,080
  float* combine = (float*)d_out;
  float* maskout = combine + per_tensor;
  float* counts = maskout + per_tensor;

  int* eidx = (int*)d_ws;
  float* gate = (float*)((char*)d_ws + S_TOKENS * sizeof(int));
  float* randv = (float*)((char*)d_ws + 2ull * S_TOKENS * sizeof(int));

  // 1) Zero the 671 MB output (the real cost; NT B128 stores).
  zero_out_kernel<<<4096, 256, 0, stream>>>((float*)d_out, (long long)out_size);

  // 2) Per-token routing: 8 tokens per 256-thread block (1 wave32 each).
  router_kernel<<<S_TOKENS / 8, 256, 0, stream>>>(x, rnd, eidx, gate, randv);

  // 3) Per-expert capacity selection + scatter of the <=8192 nonzeros.
  expert_kernel<<<N_EXPERTS, EK_THREADS, 0, stream>>>(eidx, gate, randv,
                                                      combine, maskout, counts);
}